// CompactBilinearPooling_57990648430749
// MI455X (gfx1250) — compile-verified
//
#include <hip/hip_runtime.h>
#include <hip/hip_bf16.h>

// ---------------------------------------------------------------------------
// Compact Bilinear Pooling on gfx1250 (MI455X):
//   1) count-sketch both inputs into LDS via ds_add_f32 atomics
//   2) 16384-pt FFT as four-step (128x128) -> all butterflies become
//      128x128x128 f32 GEMMs executed with V_WMMA_F32_16X16X4_F32
//   3) pointwise complex product, inverse four-step, write real part
// One workgroup (256 thr = 8 waves) per batch row; ~272 KB dynamic LDS.
// DFT matrix F (re/im) + twiddle table T (re/im) precomputed into d_ws.
// ---------------------------------------------------------------------------

typedef __attribute__((ext_vector_type(2))) float v2f;
typedef __attribute__((ext_vector_type(8))) float v8f;

#define BATCH      256
#define DIN        4096
#define NFFT       16384
#define R          128          // NFFT = R * R
#define SLD_ROW    132          // padded ld for 16x128 row staging (bank-conflict free)
#define SLD_COL    16           // ld for 128x16 column staging
#define STAGE_F    (16 * SLD_ROW)          // 2112 floats per plane (>= 128*16)
#define LDS_FLOATS (4 * NFFT + 2 * STAGE_F)
#define LDS_BYTES  (LDS_FLOATS * 4)        // 279,040 B  (< 320 KB WGP LDS)

// d_ws layout (floats)
#define WS_FR 0
#define WS_FI (NFFT)
#define WS_TR (2 * NFFT)
#define WS_TI (3 * NFFT)

// ---------------------------------------------------------------------------
// Table init: F[r][c] = W128^{rc} = cos - i sin (rc reduced mod 128),
//             T[r][c] = W16384^{rc}.
// ---------------------------------------------------------------------------
__global__ void cbp_init_tables(float* __restrict__ ws) {
  int idx = blockIdx.x * blockDim.x + threadIdx.x;  // 0..16383
  if (idx >= NFFT) return;
  int r = idx >> 7, c = idx & (R - 1);
  const float twoPi = 6.28318530717958647692f;
  int rc128 = (r * c) & (R - 1);
  float s, co;
  __sincosf(twoPi * (float)rc128 * (1.0f / 128.0f), &s, &co);
  ws[WS_FR + idx] = co;
  ws[WS_FI + idx] = -s;
  float sN, cN;
  __sincosf(twoPi * (float)(r * c) * (1.0f / 16384.0f), &sN, &cN);
  ws[WS_TR + idx] = cN;
  ws[WS_TI + idx] = -sN;
}

// ---------------------------------------------------------------------------
// WMMA helpers (f32 16x16x4).  ISA 7.12.2 layouts:
//   A 16x4 : lanes 0-15 -> M=0..15 ; VGPR0 = {K0 | K2}, VGPR1 = {K1 | K3}
//   B 4x16 : lanes hold N=0..15   ; VGPR0 = {K0 | K2}, VGPR1 = {K1 | K3}
//   C/D    : VGPR i -> row i (lanes 0-15) / row i+8 (lanes 16-31)
// f32 WMMA has no A/B NEG modifier (ISA: NEG={CNeg,0,0}) -> negate operands
// in VALU when a subtraction is needed.
// ---------------------------------------------------------------------------
__device__ __forceinline__ v8f wmma_f32(v2f a, v2f b, v8f c) {
  return __builtin_amdgcn_wmma_f32_16x16x4_f32(false, a, false, b,
                                               (short)0, c, false, false);
}

__device__ __forceinline__ v2f load_a_tile(const float* __restrict__ src,
                                           int ld, int row0, int k0, int lane) {
  int m  = lane & 15;
  int kk = (lane >> 4) << 1;       // 0 or 2
  const float* p = src + (row0 + m) * ld + (k0 + kk);
  v2f a; a.x = p[0]; a.y = p[1];
  return a;
}

__device__ __forceinline__ v2f load_b_tile(const float* __restrict__ src,
                                           int ld, int k0, int col0, int lane) {
  int n  = lane & 15;
  int kk = (lane >> 4) << 1;       // 0 or 2
  const float* p = src + (k0 + kk) * ld + (col0 + n);
  v2f b; b.x = p[0]; b.y = p[ld];
  return b;
}

__device__ __forceinline__ v2f vneg(v2f v) { v2f r; r.x = -v.x; r.y = -v.y; return r; }

// ---------------------------------------------------------------------------
// Column pass (left-multiply by F/G), in place per 16-column block.
// MODE 0: forward step1  : B = F*A (A imag == 0), fuse fwd twiddle at store
// MODE 1: inverse step3  : z_re = (1/N) * Re( conj(F) * E' ) = (Fr*Er + Fi*Ei)/N
// ---------------------------------------------------------------------------
template <int MODE>
__device__ void col_pass(float* re, float* im, float* sRe, float* sIm,
                         const float* __restrict__ Fr, const float* __restrict__ Fi,
                         const float* __restrict__ Tr, const float* __restrict__ Ti,
                         int tid, int lane, int wave) {
  for (int cb = 0; cb < 8; ++cb) {
    int c0 = cb * 16;
    for (int t = tid; t < R * 16; t += 256) {           // stage A[:, c0..c0+15]
      int rr = t >> 4, cc = t & 15;
      sRe[rr * SLD_COL + cc] = re[rr * R + c0 + cc];
      if (MODE == 1) sIm[rr * SLD_COL + cc] = im[rr * R + c0 + cc];
    }
    __syncthreads();

    int r0 = wave * 16;
    v8f accRe = {}; v8f accIm = {};
    for (int k0 = 0; k0 < R; k0 += 4) {
      v2f fr = load_a_tile(Fr, R, r0, k0, lane);
      v2f fi = load_a_tile(Fi, R, r0, k0, lane);
      v2f sr = load_b_tile(sRe, SLD_COL, k0, 0, lane);
      if (MODE == 0) {
        accRe = wmma_f32(fr, sr, accRe);                // Br = Fr*Ar
        accIm = wmma_f32(fi, sr, accIm);                // Bi = Fi*Ar
      } else {
        v2f si = load_b_tile(sIm, SLD_COL, k0, 0, lane);
        accRe = wmma_f32(fr, sr, accRe);                // Fr*Er
        accRe = wmma_f32(fi, si, accRe);                // + Fi*Ei
      }
    }

    int col = lane & 15;
    int rbase = (lane >> 4) << 3;
#pragma unroll
    for (int i = 0; i < 8; ++i) {
      int r  = r0 + rbase + i;
      int cg = c0 + col;
      if (MODE == 0) {
        float br = accRe[i], bi = accIm[i];
        float tr = Tr[r * R + cg], ti = Ti[r * R + cg]; // fwd twiddle W_N^{r*cg}
        re[r * R + cg] = br * tr - bi * ti;
        im[r * R + cg] = br * ti + bi * tr;
      } else {
        re[r * R + cg] = accRe[i] * (1.0f / 16384.0f);  // real output, 1/N
      }
    }
    __syncthreads();
  }
}

// ---------------------------------------------------------------------------
// Row pass (right-multiply by F/G), in place per 16-row block.
// MODE 0: forward step3 : D = C*F      (full complex, no twiddle)
// MODE 1: inverse step1 : E = Z*conj(F), fuse conj twiddle at store
// ---------------------------------------------------------------------------
template <int MODE>
__device__ void row_pass(float* re, float* im, float* sRe, float* sIm,
                         const float* __restrict__ Fr, const float* __restrict__ Fi,
                         const float* __restrict__ Tr, const float* __restrict__ Ti,
                         int tid, int lane, int wave) {
  for (int rb = 0; rb < 8; ++rb) {
    int r0 = rb * 16;
    for (int t = tid; t < 16 * R; t += 256) {           // stage A[r0..r0+15, :]
      int rr = t >> 7, cc = t & (R - 1);
      sRe[rr * SLD_ROW + cc] = re[(r0 + rr) * R + cc];
      sIm[rr * SLD_ROW + cc] = im[(r0 + rr) * R + cc];
    }
    __syncthreads();

    int c0 = wave * 16;
    v8f accRe = {}; v8f accIm = {};
    for (int k0 = 0; k0 < R; k0 += 4) {
      v2f sr = load_a_tile(sRe, SLD_ROW, 0, k0, lane);
      v2f si = load_a_tile(sIm, SLD_ROW, 0, k0, lane);
      v2f fr = load_b_tile(Fr, R, k0, c0, lane);
      v2f fi = load_b_tile(Fi, R, k0, c0, lane);
      if (MODE == 0) {
        accRe = wmma_f32(sr, fr, accRe);                // Sr*Fr
        accRe = wmma_f32(vneg(si), fi, accRe);          // - Si*Fi
        accIm = wmma_f32(sr, fi, accIm);                // Sr*Fi
        accIm = wmma_f32(si, fr, accIm);                // + Si*Fr
      } else {
        accRe = wmma_f32(sr, fr, accRe);                // Zr*Fr
        accRe = wmma_f32(si, fi, accRe);                // + Zi*Fi
        accIm = wmma_f32(sr, vneg(fi), accIm);          // -Zr*Fi
        accIm = wmma_f32(si, fr, accIm);                // + Zi*Fr
      }
    }

    int col = lane & 15;
    int rbase = (lane >> 4) << 3;
#pragma unroll
    for (int i = 0; i < 8; ++i) {
      int r  = r0 + rbase + i;
      int cg = c0 + col;
      if (MODE == 0) {
        re[r * R + cg] = accRe[i];
        im[r * R + cg] = accIm[i];
      } else {
        float tr = Tr[r * R + cg], ti = -Ti[r * R + cg]; // conj twiddle
        float er = accRe[i], ei = accIm[i];
        re[r * R + cg] = er * tr - ei * ti;
        im[r * R + cg] = er * ti + ei * tr;
      }
    }
    __syncthreads();
  }
}

// ---------------------------------------------------------------------------
// Main kernel: one workgroup per batch row.
// ---------------------------------------------------------------------------
extern "C" __global__ void __launch_bounds__(256, 1)
cbp_main(const float* __restrict__ x1, const float* __restrict__ x2,
         const int* __restrict__ h1, const float* __restrict__ s1,
         const float* __restrict__ ws, float* __restrict__ out) {
  extern __shared__ float lds[];
  float* y1re = lds;
  float* y1im = y1re + NFFT;
  float* y2re = y1im + NFFT;
  float* y2im = y2re + NFFT;
  float* sRe  = y2im + NFFT;     // STAGE_F floats
  float* sIm  = sRe + STAGE_F;

  const float* Fr = ws + WS_FR;
  const float* Fi = ws + WS_FI;
  const float* Tr = ws + WS_TR;
  const float* Ti = ws + WS_TI;

  int tid  = threadIdx.x;
  int lane = tid & 31;           // wave32
  int wave = tid >> 5;           // 8 waves
  int b    = blockIdx.x;

  // ---- zero sketch buffers -------------------------------------------------
  for (int t = tid; t < 4 * NFFT; t += 256) lds[t] = 0.0f;
  __syncthreads();

  // ---- count-sketch scatter (ds_add_f32 atomics) ---------------------------
  const float* x1r = x1 + (size_t)b * DIN;
  const float* x2r = x2 + (size_t)b * DIN;
  for (int d = tid; d < DIN; d += 256) {
    int   h = h1[d];
    float s = s1[d];
    atomicAdd(&y1re[h], s * x1r[d]);
    atomicAdd(&y2re[h], s * x2r[d]);
  }
  __syncthreads();

  // ---- forward four-step FFT of both sketches (in place) -------------------
  col_pass<0>(y1re, y1im, sRe, sIm, Fr, Fi, Tr, Ti, tid, lane, wave);
  row_pass<0>(y1re, y1im, sRe, sIm, Fr, Fi, Tr, Ti, tid, lane, wave);
  col_pass<0>(y2re, y2im, sRe, sIm, Fr, Fi, Tr, Ti, tid, lane, wave);
  row_pass<0>(y2re, y2im, sRe, sIm, Fr, Fi, Tr, Ti, tid, lane, wave);

  // ---- pointwise spectral product Z = Y1 .* Y2 (same digit-scramble) -------
  for (int t = tid; t < NFFT; t += 256) {
    float ar = y1re[t], ai = y1im[t];
    float br = y2re[t], bi = y2im[t];
    y1re[t] = ar * br - ai * bi;
    y1im[t] = ar * bi + ai * br;
  }
  __syncthreads();

  // ---- inverse four-step FFT (row pass first on scrambled layout) ----------
  row_pass<1>(y1re, y1im, sRe, sIm, Fr, Fi, Tr, Ti, tid, lane, wave);
  col_pass<1>(y1re, y1im, sRe, sIm, Fr, Fi, Tr, Ti, tid, lane, wave);

  // ---- write real result row (vectorized b128 stores) ----------------------
  float4* dst = (float4*)(out + (size_t)b * NFFT);
  const float4* src = (const float4*)y1re;
  for (int t = tid; t < NFFT / 4; t += 256) dst[t] = src[t];
}

// ---------------------------------------------------------------------------
// Host entry
// ---------------------------------------------------------------------------
extern "C" void kernel_launch(void* const* d_in, const int* in_sizes, int n_in,
                              void* d_out, int out_size, void* d_ws, size_t ws_size,
                              hipStream_t stream) {
  const float* x1 = (const float*)d_in[0];
  const float* x2 = (const float*)d_in[1];
  const int*   h1 = (const int*)d_in[2];
  const float* s1 = (const float*)d_in[3];
  // d_in[4] = output_size scalar (compile-time 16384 here)

  float* ws  = (float*)d_ws;   // 4 * 16384 floats = 256 KB of tables
  float* out = (float*)d_out;

  int B = in_sizes[0] / DIN;   // 256

  // Allow >64KB dynamic LDS (best effort; ROCm honors device WGP limit).
  (void)hipFuncSetAttribute((const void*)cbp_main,
                            hipFuncAttributeMaxDynamicSharedMemorySize,
                            LDS_BYTES);

  cbp_init_tables<<<(NFFT + 255) / 256, 256, 0, stream>>>(ws);
  cbp_main<<<B, 256, LDS_BYTES, stream>>>(x1, x2, h1, s1, ws, out);
}